// Cache3DPCD_6399501271057
// MI455X (gfx1250) — compile-verified
//
#include <hip/hip_runtime.h>
#include <stdint.h>

// ---------------------------------------------------------------------------
// Point-splat renderer for MI455X (gfx1250, wave32).
// Pipeline (all intermediates sized to live in the 192MB L2):
//   k_setup      : fuse source->target rigid transforms into 3x4 matrices
//   k_unproject  : source pixels -> cam-space points + colors (AoS, 32B/pt)
//   per target:
//     k_project<false> : WMMA-batched transform, count frags/pixel (atomics)
//     k_scan           : exclusive scan of per-pixel counts (1 block)
//     k_project<true>  : WMMA-batched transform, scatter 64-bit keyed frags
//     k_composite      : wave-per-pixel top-32 extraction + alpha compositing
// ---------------------------------------------------------------------------

typedef float v2f __attribute__((ext_vector_type(2)));
typedef float v8f __attribute__((ext_vector_type(8)));

constexpr int   Nn = 4, Tn = 4, Hh = 256, Ww = 256;
constexpr int   HW = Hh * Ww;            // 65536
constexpr int   Pt = Nn * HW;            // 262144 points
constexpr int   FRAG_CAP = 9 * Pt;       // worst-case frags per target
constexpr float R2f   = 0.01f * 0.01f;   // RADIUS^2
constexpr float R2inv = 1.0f / R2f;
constexpr float S2f   = (2.0f / 256.0f) * (2.0f / 256.0f);

__device__ __forceinline__ int lane_id() { return (int)(threadIdx.x & 31u); }

// ---------------------------------------------------------------------------
// Wave-cooperative rigid transform of 32 points via V_WMMA_F32_16X16X4_F32.
// A (16x4): rows 0..2 = transform rows over K = [x y z 1].  Layout per ISA:
//   lanes 0-15 hold row=lane {K0,K1}; lanes 16-31 hold row=lane-16 {K2,K3}.
// B (4x16): column n = homogeneous point n.
// D (16x16 f32): vgpr r, lane L -> row r+(L>>4)*8, col L&15. Rows 0..2 of
// point n land in D[0..2] at lane n (rows 8..10 are zero), so one shuffle
// per component redistributes results back to the owning lane.
// Requires EXEC == all ones (callers are branch-free up to this point).
// ---------------------------------------------------------------------------
__device__ __forceinline__ void wave_transform_wmma(const float* __restrict__ M,
    float x, float y, float z, float& X, float& Y, float& Z)
{
  const int lane = lane_id();
  const int hf   = lane >> 4;   // which K-pair half this lane feeds
  const int idx  = lane & 15;

  float a0 = (idx < 3) ? (hf ? M[idx * 4 + 2] : M[idx * 4 + 0]) : 0.0f;
  float a1 = (idx < 3) ? (hf ? M[idx * 4 + 3] : M[idx * 4 + 1]) : 0.0f;
  v2f A; A[0] = a0; A[1] = a1;
  v8f Cz = {0.f, 0.f, 0.f, 0.f, 0.f, 0.f, 0.f, 0.f};

  // Points 0..15 as B
  float x0 = __shfl(x, idx), y0 = __shfl(y, idx), z0 = __shfl(z, idx);
  v2f B0; B0[0] = hf ? z0 : x0; B0[1] = hf ? 1.0f : y0;
  v8f D0 = __builtin_amdgcn_wmma_f32_16x16x4_f32(false, A, false, B0,
                                                 (short)0, Cz, false, false);
  // Points 16..31 as B
  float x1 = __shfl(x, 16 + idx), y1 = __shfl(y, 16 + idx), z1 = __shfl(z, 16 + idx);
  v2f B1; B1[0] = hf ? z1 : x1; B1[1] = hf ? 1.0f : y1;
  v8f D1 = __builtin_amdgcn_wmma_f32_16x16x4_f32(false, A, false, B1,
                                                 (short)0, Cz, false, false);

  float Xa = __shfl(D0[0], idx), Xb = __shfl(D1[0], idx);
  float Ya = __shfl(D0[1], idx), Yb = __shfl(D1[1], idx);
  float Za = __shfl(D0[2], idx), Zb = __shfl(D1[2], idx);
  X = hf ? Xb : Xa;
  Y = hf ? Yb : Ya;
  Z = hf ? Zb : Za;
}

// ---------------------------------------------------------------------------
// Fused transforms: cam_t = Rt*Rs^T*cam_s + (tt - Rt*Rs^T*ts), one 3x4 per
// (target, source) pair, row-major 12 floats.
// ---------------------------------------------------------------------------
__global__ void k_setup(const float* __restrict__ Es, const float* __restrict__ Et,
                        float* __restrict__ Mts)
{
  int i = (int)threadIdx.x;
  if (i >= Tn * Nn) return;
  int t = i / Nn, s = i % Nn;
  const float* es = Es + s * 16;
  const float* et = Et + t * 16;
  float M[12];
  for (int r = 0; r < 3; ++r) {
    for (int k = 0; k < 3; ++k) {
      float acc = 0.f;
      for (int j = 0; j < 3; ++j) acc += et[r * 4 + j] * es[k * 4 + j];
      M[r * 4 + k] = acc;
    }
    float b = et[r * 4 + 3];
    for (int k = 0; k < 3; ++k) b -= M[r * 4 + k] * es[k * 4 + 3];
    M[r * 4 + 3] = b;
  }
  for (int j = 0; j < 12; ++j) Mts[i * 12 + j] = M[j];
}

// ---------------------------------------------------------------------------
// Unproject: cam-space point (x,y,z) + color (r,g,b) per source pixel,
// packed 8 floats (two 128-bit stores).
// ---------------------------------------------------------------------------
__global__ void k_unproject(const float* __restrict__ img,
                            const float* __restrict__ dep,
                            const float* __restrict__ Kin,
                            float* __restrict__ pc)
{
  int p   = blockIdx.x * blockDim.x + threadIdx.x;   // < Pt exactly
  int n   = p >> 16;
  int rem = p & (HW - 1);
  int y   = rem >> 8, x = rem & 255;
  float z  = dep[n * HW + rem];
  float fx = Kin[n * 9 + 0], fy = Kin[n * 9 + 4];
  float cx = Kin[n * 9 + 2], cy = Kin[n * 9 + 5];
  float cxm = ((float)x + 0.5f - cx) / fx * z;
  float cym = ((float)y + 0.5f - cy) / fy * z;
  float4 v0 = make_float4(cxm, cym, z, img[(n * 3 + 0) * HW + rem]);
  float4 v1 = make_float4(img[(n * 3 + 1) * HW + rem],
                          img[(n * 3 + 2) * HW + rem], 0.f, 0.f);
  float4* o = (float4*)(pc + (size_t)p * 8);
  o[0] = v0; o[1] = v1;
}

// ---------------------------------------------------------------------------
// Project + splat.  WRITE=false: count frags per pixel.  WRITE=true: scatter
// 64-bit keys (z_bits<<32 | off*Pt+p) -> exact reference stable ordering.
// Each wave covers 32 consecutive points -> source view uniform per wave.
// Single reciprocal (1/zc) feeds both u and v: one IEEE div expansion
// instead of two on the hot path.
// ---------------------------------------------------------------------------
template <bool WRITE>
__global__ void k_project(const float* __restrict__ pc,
                          const float* __restrict__ Mts,
                          const float* __restrict__ TK, int t,
                          unsigned* __restrict__ cnt_or_cur,
                          unsigned long long* __restrict__ frags)
{
  int p = blockIdx.x * blockDim.x + threadIdx.x;     // < Pt exactly
  int s = p >> 16;
  const float* M = Mts + (t * Nn + s) * 12;
  const float4 q0 = ((const float4*)(pc + (size_t)p * 8))[0];

  float X, Y, Z;
  wave_transform_wmma(M, q0.x, q0.y, q0.z, X, Y, Z);

  float fx = TK[t * 9 + 0], fy = TK[t * 9 + 4];
  float cx = TK[t * 9 + 2], cy = TK[t * 9 + 5];
  float zc   = fmaxf(Z, 1e-6f);
  float invz = 1.0f / zc;
  float u  = fx * X * invz + cx;
  float v  = fy * Y * invz + cy;
  int bx = (int)floorf(u);
  int by = (int)floorf(v);
  bool zok = Z > 1e-6f;
  unsigned zbits = __float_as_uint(Z);   // positive -> monotone sortable

#pragma unroll
  for (int oy = 0; oy < 3; ++oy) {
#pragma unroll
    for (int ox = 0; ox < 3; ++ox) {
      int px = bx + ox - 1, py = by + oy - 1;
      float du = u - ((float)px + 0.5f);
      float dv = v - ((float)py + 0.5f);
      float d2 = (du * du + dv * dv) * S2f;
      bool ok = zok && (px >= 0) && (px < Ww) && (py >= 0) && (py < Hh) &&
                (d2 <= R2f);
      if (ok) {
        int pid = py * Ww + px;
        if (WRITE) {
          unsigned pos = atomicAdd(&cnt_or_cur[pid], 1u);
          unsigned long long key =
              ((unsigned long long)zbits << 32) |
              (unsigned)((oy * 3 + ox) * Pt + p);
          frags[pos] = key;
        } else {
          atomicAdd(&cnt_or_cur[pid], 1u);
        }
      }
    }
  }
}

// ---------------------------------------------------------------------------
// Exclusive scan of 65536 counts, single block (1024 thr x 64 serial each).
// ---------------------------------------------------------------------------
__global__ void k_scan(const unsigned* __restrict__ cnt,
                       unsigned* __restrict__ offs,
                       unsigned* __restrict__ cur)
{
  __shared__ unsigned sm[1024];
  const int tid  = (int)threadIdx.x;
  const int base = tid * 64;
  unsigned s = 0;
  for (int i = 0; i < 64; ++i) s += cnt[base + i];
  sm[tid] = s;
  __syncthreads();
  for (int d = 1; d < 1024; d <<= 1) {
    unsigned vv = (tid >= d) ? sm[tid - d] : 0u;
    __syncthreads();
    sm[tid] += vv;
    __syncthreads();
  }
  unsigned run = (tid > 0) ? sm[tid - 1] : 0u;
  for (int i = 0; i < 64; ++i) {
    unsigned c = cnt[base + i];
    offs[base + i] = run;
    cur[base + i]  = run;
    run += c;
  }
}

// ---------------------------------------------------------------------------
// Composite: one wave32 per pixel.  Stage the fragment segment into LDS,
// then extract the 32 smallest keys (strictly increasing, keys unique) and
// alpha-composite front-to-back.  All lanes hold the reduced key, so alpha /
// color recompute is broadcast work; lane 0 stores the pixel.
// ---------------------------------------------------------------------------
constexpr int LCAP = 512;

__global__ __launch_bounds__(128) void k_composite(
    const unsigned long long* __restrict__ frags,
    const unsigned* __restrict__ offs, const unsigned* __restrict__ cur,
    const float* __restrict__ pc, const float* __restrict__ Mts,
    const float* __restrict__ TK, int t, float* __restrict__ out)
{
  __shared__ unsigned long long lf[4][LCAP];
  const int w    = (int)(threadIdx.x >> 5);
  const int lane = (int)(threadIdx.x & 31u);
  const int pid  = blockIdx.x * 4 + w;

  const unsigned start = offs[pid];
  const unsigned end   = cur[pid];
  const int n = (int)(end - start);
  __builtin_prefetch(frags + start, 0, 1);            // global_prefetch_b8

  const int nc = n < LCAP ? n : LCAP;
  for (int i = lane; i < nc; i += 32) lf[w][i] = frags[start + i];
  __syncthreads();

  const float fx = TK[t * 9 + 0], fy = TK[t * 9 + 4];
  const float cx = TK[t * 9 + 2], cy = TK[t * 9 + 5];
  const int px = pid & (Ww - 1), py = pid >> 8;

  float Tr = 1.0f, oR = 0.f, oG = 0.f, oB = 0.f;
  unsigned long long last = 0ull;
  const int m = n < 32 ? n : 32;

  for (int k = 0; k < m; ++k) {
    unsigned long long best = ~0ull;
    for (int i = lane; i < nc; i += 32) {
      unsigned long long f = lf[w][i];
      if (f > last && f < best) best = f;
    }
    for (int i = LCAP + lane; i < n; i += 32) {      // rare overflow tail
      unsigned long long f = frags[start + i];
      if (f > last && f < best) best = f;
    }
#pragma unroll
    for (int d = 16; d > 0; d >>= 1) {
      unsigned long long o = __shfl_xor(best, d);
      best = best < o ? best : o;
    }
    last = best;

    unsigned idx = (unsigned)(best & 0xffffffffull);
    unsigned off = idx / (unsigned)Pt;
    unsigned p   = idx - off * (unsigned)Pt;
    unsigned s   = p >> 16;
    const float* M = Mts + (t * Nn + (int)s) * 12;
    const float4 q0 = ((const float4*)(pc + (size_t)p * 8))[0];
    const float4 q1 = ((const float4*)(pc + (size_t)p * 8))[1];
    float X = M[0] * q0.x + M[1] * q0.y + M[2]  * q0.z + M[3];
    float Y = M[4] * q0.x + M[5] * q0.y + M[6]  * q0.z + M[7];
    float Z = M[8] * q0.x + M[9] * q0.y + M[10] * q0.z + M[11];
    float zc   = fmaxf(Z, 1e-6f);
    float invz = 1.0f / zc;
    float u  = fx * X * invz + cx;
    float v  = fy * Y * invz + cy;
    float du = u - ((float)px + 0.5f);
    float dv = v - ((float)py + 0.5f);
    float d2 = (du * du + dv * dv) * S2f;
    float a  = 1.0f - d2 * R2inv;
    float wA = a * Tr;
    oR += wA * q0.w;
    oG += wA * q1.x;
    oB += wA * q1.y;
    Tr *= (1.0f - a);
  }

  if (lane == 0) {
    float* o = out + ((size_t)(t * HW) + (size_t)pid) * 3;
    o[0] = oR; o[1] = oG; o[2] = oB;
  }
}

// ---------------------------------------------------------------------------
extern "C" void kernel_launch(void* const* d_in, const int* in_sizes, int n_in,
                              void* d_out, int out_size, void* d_ws, size_t ws_size,
                              hipStream_t stream)
{
  (void)in_sizes; (void)n_in; (void)out_size; (void)ws_size;
  const float* img = (const float*)d_in[0];   // (1,4,3,256,256)
  const float* dep = (const float*)d_in[1];   // (1,4,1,256,256)
  const float* Es  = (const float*)d_in[2];   // (1,4,4,4)
  const float* Kin = (const float*)d_in[3];   // (1,4,3,3)
  const float* Et  = (const float*)d_in[4];   // (1,4,4,4)
  const float* TK  = (const float*)d_in[5];   // (1,4,3,3)
  float* out = (float*)d_out;                 // (1,4,256,256,3)

  char*  ws = (char*)d_ws;
  size_t o  = 0;
  auto alloc = [&](size_t bytes) -> void* {
    void* p = ws + o;
    o = (o + bytes + 255) & ~(size_t)255;
    return p;
  };
  float*    Mts  = (float*)alloc((size_t)Tn * Nn * 12 * sizeof(float));
  float*    pc   = (float*)alloc((size_t)Pt * 8 * sizeof(float));
  unsigned* cnt  = (unsigned*)alloc((size_t)HW * sizeof(unsigned));
  unsigned* offs = (unsigned*)alloc((size_t)HW * sizeof(unsigned));
  unsigned* cur  = (unsigned*)alloc((size_t)HW * sizeof(unsigned));
  unsigned long long* frags =
      (unsigned long long*)alloc((size_t)FRAG_CAP * sizeof(unsigned long long));

  hipLaunchKernelGGL(k_setup, dim3(1), dim3(32), 0, stream, Es, Et, Mts);
  hipLaunchKernelGGL(k_unproject, dim3(Pt / 256), dim3(256), 0, stream,
                     img, dep, Kin, pc);

  for (int t = 0; t < Tn; ++t) {
    hipMemsetAsync(cnt, 0, (size_t)HW * sizeof(unsigned), stream);
    hipLaunchKernelGGL((k_project<false>), dim3(Pt / 256), dim3(256), 0, stream,
                       pc, Mts, TK, t, cnt, (unsigned long long*)nullptr);
    hipLaunchKernelGGL(k_scan, dim3(1), dim3(1024), 0, stream, cnt, offs, cur);
    hipLaunchKernelGGL((k_project<true>), dim3(Pt / 256), dim3(256), 0, stream,
                       pc, Mts, TK, t, cur, frags);
    hipLaunchKernelGGL(k_composite, dim3(HW / 4), dim3(128), 0, stream,
                       frags, offs, cur, pc, Mts, TK, t, out);
  }
}